// HausdorffDist_3006477107868
// MI455X (gfx1250) — compile-verified
//
#include <hip/hip_runtime.h>
#include <hip/hip_bf16.h>

typedef __attribute__((ext_vector_type(2))) float v2f;
typedef __attribute__((ext_vector_type(8))) float v8f;

#define BATCH 16
#define NPTS  4096   // adv points per batch
#define MPTS  4096   // ori points per batch
#define NTILES (MPTS / 16)
#define ROWS_PER_BLOCK 128   // 8 waves * 16 rows
#define THREADS 256

// Monotone float <-> uint key mapping (order-preserving), so native
// u32 atomicMin/atomicMax (ds_min_u32 / global_atomic_min_u32) implement
// float min/max deterministically.
__device__ __forceinline__ unsigned f2key(float f) {
    unsigned b = __float_as_uint(f);
    return (b & 0x80000000u) ? ~b : (b | 0x80000000u);
}
__device__ __forceinline__ float key2f(unsigned k) {
    return (k & 0x80000000u) ? __uint_as_float(k & 0x7FFFFFFFu)
                             : __uint_as_float(~k);
}

__global__ void hausdorff_init_kernel(unsigned* __restrict__ colmin_g,
                                      unsigned* __restrict__ rowmax_g) {
    int i = blockIdx.x * THREADS + threadIdx.x;
    if (i < BATCH * MPTS) colmin_g[i] = 0xFFFFFFFFu;  // +inf key for min
    if (i < BATCH)        rowmax_g[i] = 0u;           // -inf key for max
}

// One block = (batch b, 128 adv rows). 8 waves, each wave owns 16 rows and
// sweeps all 4096 ori columns in 16-wide tiles via V_WMMA_F32_16X16X4_F32.
// D = A x B + C with A row = (x0,x1,x2,|x|^2), B col = (-2y0,-2y1,-2y2,1),
// C = |y|^2  ==>  D[m][n] = squared distance, exact f32.
__global__ __launch_bounds__(THREADS)
void hausdorff_tiles_kernel(const float* __restrict__ adv,
                            const float* __restrict__ ori,
                            unsigned* __restrict__ colmin_g,
                            unsigned* __restrict__ rowmax_g) {
    __shared__ float4   oriS[MPTS];      // (-2y0, -2y1, -2y2, |y|^2)  64 KB
    __shared__ unsigned colminS[MPTS];   // per-block col-min keys     16 KB

    const int b   = blockIdx.y;
    const int tid = threadIdx.x;

    // Stage & preprocess ori[b] into LDS once per block.
    const float* orib = ori + (size_t)b * MPTS * 3;
    for (int i = tid; i < MPTS; i += THREADS) {
        float y0 = orib[i * 3 + 0];
        float y1 = orib[i * 3 + 1];
        float y2 = orib[i * 3 + 2];
        oriS[i] = make_float4(-2.0f * y0, -2.0f * y1, -2.0f * y2,
                              y0 * y0 + y1 * y1 + y2 * y2);
        colminS[i] = 0xFFFFFFFFu;
    }
    __syncthreads();

    const int lane = tid & 31;
    const int wave = tid >> 5;
    const int half = lane >> 4;   // 0: lanes 0-15 (K0,K1) ; 1: lanes 16-31 (K2,K3)
    const int l16  = lane & 15;

    // A-matrix (16x4 f32): row = this wave's adv point; K3 column = |x|^2.
    const int    r  = blockIdx.x * ROWS_PER_BLOCK + wave * 16 + l16;
    const float* ap = adv + ((size_t)b * NPTS + r) * 3;
    const float  x0 = ap[0], x1 = ap[1], x2 = ap[2];
    v2f a;
    a[0] = half ? x2 : x0;
    a[1] = half ? (x0 * x0 + x1 * x1 + x2 * x2) : x1;

    const float INF = __builtin_inff();
    v8f rowmin;
#pragma unroll
    for (int i = 0; i < 8; ++i) rowmin[i] = INF;

    // Software-pipelined over column tiles: prefetch next ori float4
    // (branchless wrap index) so ds_load latency hides under WMMA + mins.
    float4 o = oriS[l16];
#pragma unroll 4
    for (int ct = 0; ct < NTILES; ++ct) {
        const float4 oNext = oriS[((ct + 1) & (NTILES - 1)) * 16 + l16];

        // B-matrix (4x16 f32): col = ori point; K3 row = 1.0.
        v2f bm;
        bm[0] = half ? o.z : o.x;
        bm[1] = half ? 1.0f : o.y;

        // C-matrix: |y_col|^2 broadcast down the 8 row-VGPRs.
        v8f cm;
#pragma unroll
        for (int i = 0; i < 8; ++i) cm[i] = o.w;

        v8f d = __builtin_amdgcn_wmma_f32_16x16x4_f32(
            false, a, false, bm, (short)0, cm, false, false);

        // Running min over ori (columns) for this wave's 16 rows.
#pragma unroll
        for (int i = 0; i < 8; ++i) rowmin[i] = fminf(rowmin[i], d[i]);

        // Column min over this lane-half's 8 rows; both halves fire-and-forget
        // ds_min_u32 into the same slot (covers all 16 rows, no shuffle, no
        // divergence, no wait).
        float cmin = fminf(fminf(fminf(d[0], d[1]), fminf(d[2], d[3])),
                           fminf(fminf(d[4], d[5]), fminf(d[6], d[7])));
        atomicMin(&colminS[ct * 16 + l16], f2key(cmin));

        o = oNext;
    }

    // Reduce row mins across the 16 lanes of each half (same row, diff cols).
    float wmax = -INF;
#pragma unroll
    for (int i = 0; i < 8; ++i) {
        float v = rowmin[i];
        v = fminf(v, __shfl_xor(v, 1, 16));
        v = fminf(v, __shfl_xor(v, 2, 16));
        v = fminf(v, __shfl_xor(v, 4, 16));
        v = fminf(v, __shfl_xor(v, 8, 16));
        wmax = fmaxf(wmax, v);   // max over this wave's rows (dist_xy -> max_n)
    }
    wmax = fmaxf(wmax, __shfl_xor(wmax, 16, 32));
    if (lane == 0) atomicMax(&rowmax_g[b], f2key(wmax));

    __syncthreads();
    // Flush block-level column mins to the global per-batch col-min array
    // (fire-and-forget global_atomic_min_u32).
    for (int i = tid; i < MPTS; i += THREADS)
        atomicMin(&colmin_g[(size_t)b * MPTS + i], colminS[i]);
}

// Single block: per-batch max over col-mins, combine with row-max, fixed-order
// sum -> mean. Fully deterministic.
__global__ __launch_bounds__(THREADS)
void hausdorff_finalize_kernel(const unsigned* __restrict__ colmin_g,
                               const unsigned* __restrict__ rowmax_g,
                               float* __restrict__ out) {
    __shared__ unsigned red[THREADS];
    __shared__ float    losses[BATCH];
    const int tid = threadIdx.x;

    for (int b = 0; b < BATCH; ++b) {
        unsigned k = 0u;  // -inf key
        for (int i = tid; i < MPTS; i += THREADS) {
            unsigned v = colmin_g[(size_t)b * MPTS + i];
            k = (v > k) ? v : k;   // max over keys == key of max value
        }
        red[tid] = k;
        __syncthreads();
        for (int s = THREADS / 2; s > 0; s >>= 1) {
            if (tid < s) {
                unsigned v = red[tid + s];
                if (v > red[tid]) red[tid] = v;
            }
            __syncthreads();
        }
        if (tid == 0) {
            float colmax = key2f(red[0]);        // max_m dist_yx
            float rowmax = key2f(rowmax_g[b]);   // max_n dist_xy
            losses[b] = fmaxf(rowmax, colmax);
        }
        __syncthreads();
    }
    if (tid == 0) {
        float s = 0.0f;
        for (int b = 0; b < BATCH; ++b) s += losses[b];
        out[0] = s * (1.0f / (float)BATCH);
    }
}

extern "C" void kernel_launch(void* const* d_in, const int* in_sizes, int n_in,
                              void* d_out, int out_size, void* d_ws, size_t ws_size,
                              hipStream_t stream) {
    const float* adv = (const float*)d_in[0];   // [16, 4096, 3] f32
    const float* ori = (const float*)d_in[1];   // [16, 4096, 3] f32
    float* out = (float*)d_out;                 // scalar f32

    unsigned* ws_u     = (unsigned*)d_ws;
    unsigned* colmin_g = ws_u;                  // BATCH*MPTS u32 keys (256 KB)
    unsigned* rowmax_g = ws_u + BATCH * MPTS;   // BATCH u32 keys

    const int init_elems  = BATCH * MPTS;
    const int init_blocks = (init_elems + THREADS - 1) / THREADS;
    hausdorff_init_kernel<<<init_blocks, THREADS, 0, stream>>>(colmin_g, rowmax_g);

    dim3 grid(NPTS / ROWS_PER_BLOCK, BATCH);    // (32, 16)
    hausdorff_tiles_kernel<<<grid, THREADS, 0, stream>>>(adv, ori, colmin_g, rowmax_g);

    hausdorff_finalize_kernel<<<1, THREADS, 0, stream>>>(colmin_g, rowmax_g, out);
}